// HRWKV_7_56684978372998
// MI455X (gfx1250) — compile-verified
//
#include <hip/hip_runtime.h>

typedef float v2f __attribute__((ext_vector_type(2)));
typedef float v8f __attribute__((ext_vector_type(8)));

#define Bsz 64
#define Hn  32
#define Kd  128
#define Vd  128

__global__ __launch_bounds__(256) void rwkv7_dplr_t1_kernel(
    const float* __restrict__ q,  const float* __restrict__ kk,
    const float* __restrict__ vv, const float* __restrict__ aa,
    const float* __restrict__ bb, const float* __restrict__ gk,
    const float* __restrict__ h0, float* __restrict__ o_out,
    float* __restrict__ ht_out)
{
    __shared__ float Lq[Kd], Lk[Kd], Lv[Vd], La[Kd], Lb[Kd], Lg[Kd], Lqg[Kd];
    __shared__ float Ls[Vd];
    __shared__ float sp[2][Vd], tp[2][Vd];
    __shared__ float redA[Kd], redB[Kd];
    __shared__ float s_alpha, s_beta;

    const int bh  = blockIdx.x;            // 0 .. B*H-1
    const int tid = threadIdx.x;           // 0 .. 255
    const float* __restrict__ hbase = h0    + (size_t)bh * (Kd * Vd);
    float*       __restrict__ htb   = ht_out + (size_t)bh * (Kd * Vd);

    // ---- load per-(b,h) vectors, build g = exp(gk), q*g, alpha/beta partials
    if (tid < Kd) {
        float qv  = q [bh * Kd + tid];
        float kv  = kk[bh * Kd + tid];
        float av  = aa[bh * Kd + tid];
        float bv  = bb[bh * Kd + tid];
        float gv  = __expf(gk[bh * Kd + tid]);
        float vxl = vv[bh * Vd + tid];
        Lq[tid] = qv; Lk[tid] = kv; La[tid] = av; Lb[tid] = bv;
        Lg[tid] = gv; Lv[tid] = vxl; Lqg[tid] = qv * gv;
        redA[tid] = qv * bv;    // -> alpha = q.b
        redB[tid] = qv * kv;    // -> beta  = q.k
    }
    __syncthreads();

    // deterministic tree reduction over 128 elements
    for (int off = 64; off > 0; off >>= 1) {
        if (tid < off) { redA[tid] += redA[tid + off]; redB[tid] += redB[tid + off]; }
        __syncthreads();
    }
    if (tid == 0) { s_alpha = redA[0]; s_beta = redB[0]; }

    // ---- pass 1: s[v] = sum_k a[k]*H0[k,v], t[v] = sum_k (q*g)[k]*H0[k,v]
    {
        const int vcol = tid & (Vd - 1);
        const int half = tid >> 7;          // 0: k=0..63, 1: k=64..127
        const int kb   = half * 64;
        float accs = 0.f, acct = 0.f;
        #pragma unroll 8
        for (int j = 0; j < 64; ++j) {
            float hv = hbase[(kb + j) * Vd + vcol];   // int offsets -> saddr form
            accs += La [kb + j] * hv;
            acct += Lqg[kb + j] * hv;
        }
        sp[half][vcol] = accs;
        tp[half][vcol] = acct;
    }
    __syncthreads();

    if (tid < Vd) {
        float sv = sp[0][tid] + sp[1][tid];
        float tv = tp[0][tid] + tp[1][tid];
        Ls[tid] = sv;
        o_out[bh * Vd + tid] = tv + s_alpha * sv + s_beta * Lv[tid];
    }
    __syncthreads();

    // ---- pass 2: Ht = diag(g)*H0 + b s^T + k v^T, one WMMA per 16x16 tile.
    // A(16x4) = [b | k | 0 | 0],  B(4x16) = [s ; v ; x ; x],  C = g.*H0 tile.
    // A's K=2,3 columns (lanes 16-31) are zeroed via mask-multiply, which makes
    // B's rows 2,3 (lanes 16-31) dead -> B needs NO masking, so all LDS reads
    // below are unconditional (no exec predication in the hot loop).
    const int   wave = tid >> 5;                   // 0..7 -> row-tile ki
    const int   lane = tid & 31;
    const int   k0   = wave * 16;
    const int   lrow = lane & 15;
    const float msk  = (lane < 16) ? 1.0f : 0.0f;

    v2f Afrag;
    Afrag.x = msk * Lb[k0 + lrow];                 // K=0 col = b   (K=2 -> 0)
    Afrag.y = msk * Lk[k0 + lrow];                 // K=1 col = k   (K=3 -> 0)

    const int rbase = k0 + ((lane >> 4) << 3);     // C/D rows: j or j+8
    float gr[8];
    #pragma unroll
    for (int j = 0; j < 8; ++j) gr[j] = Lg[rbase + j];

    #pragma unroll
    for (int vi = 0; vi < 8; ++vi) {
        const int v0  = vi * 16;
        const int col = v0 + lrow;

        v2f Bfrag;
        Bfrag.x = Ls[v0 + lrow];                   // row K=0 = s (rows 2,3 dead)
        Bfrag.y = Lv[v0 + lrow];                   // row K=1 = v

        v8f C;
        #pragma unroll
        for (int j = 0; j < 8; ++j)
            C[j] = gr[j] * hbase[(rbase + j) * Vd + col];   // L2-hot reread

        v8f D = __builtin_amdgcn_wmma_f32_16x16x4_f32(
            /*neg_a=*/false, Afrag, /*neg_b=*/false, Bfrag,
            /*c_mod=*/(short)0, C, /*reuse_a=*/false, /*reuse_b=*/false);

        #pragma unroll
        for (int j = 0; j < 8; ++j)
            htb[(rbase + j) * Vd + col] = D[j];
    }
}

extern "C" void kernel_launch(void* const* d_in, const int* in_sizes, int n_in,
                              void* d_out, int out_size, void* d_ws, size_t ws_size,
                              hipStream_t stream) {
    (void)in_sizes; (void)n_in; (void)d_ws; (void)ws_size; (void)out_size;
    const float* q  = (const float*)d_in[0];
    const float* k  = (const float*)d_in[1];
    const float* v  = (const float*)d_in[2];
    const float* a  = (const float*)d_in[3];
    const float* b  = (const float*)d_in[4];
    const float* gk = (const float*)d_in[5];
    const float* h0 = (const float*)d_in[6];

    float* o  = (float*)d_out;                              // [B,1,H,V]
    float* ht = (float*)d_out + (size_t)Bsz * Hn * Vd;      // [B,H,K,V]

    rwkv7_dplr_t1_kernel<<<dim3(Bsz * Hn), dim3(256), 0, stream>>>(
        q, k, v, a, b, gk, h0, o, ht);
}